// RMSELoss_91164975825341
// MI455X (gfx1250) — compile-verified
//
#include <hip/hip_runtime.h>
#include <math.h>

// ---------------------------------------------------------------------------
// Masked RMSE over predictions/actuals [B=1941, S=16384] fp32, memory-bound.
// HBM roofline: ~254 MB single-pass read @ 23.3 TB/s => ~11 us floor.
// Strategy: prep gather -> tiled streaming partial sums (NT b128 loads,
// wave reduction via V_WMMA_F32_16X16X4_F32 with ones-B) -> final reduce.
// All reductions are fixed-order => deterministic across graph replays.
// ---------------------------------------------------------------------------

typedef float v4f __attribute__((ext_vector_type(4)));
typedef int   v4i __attribute__((ext_vector_type(4)));
typedef float v2f __attribute__((ext_vector_type(2)));
typedef float v8f __attribute__((ext_vector_type(8)));

#define B_ROWS     1941
#define S_COLS     16384
#define TPB        256
#define GX         (S_COLS / (TPB * 4))                       // 16 column blocks
#define ROW_CHUNKS 128
#define ROWS_PER   ((B_ROWS + ROW_CHUNKS - 1) / ROW_CHUNKS)   // 16 rows / chunk
#define NPART      (GX * ROW_CHUNKS)                          // 2048 partials

// --- Kernel 1: hoist the double-gather: starts[s] = rvsi[indexes[s]] -------
__global__ __launch_bounds__(TPB) void rmse_prep(const int* __restrict__ idx,
                                                 const int* __restrict__ rvsi,
                                                 int* __restrict__ starts) {
    int s = blockIdx.x * TPB + threadIdx.x;
    if (s < S_COLS) starts[s] = rvsi[idx[s]];
}

// --- Kernel 2: streaming masked SSE partials -------------------------------
__global__ __launch_bounds__(TPB) void rmse_partial(const float* __restrict__ pred,
                                                    const float* __restrict__ act,
                                                    const int*   __restrict__ starts,
                                                    float*       __restrict__ partials) {
    const int tid = threadIdx.x;
    const int s4  = (blockIdx.x * TPB + tid) * 4;   // 4 consecutive series cols
    const int r0  = blockIdx.y * ROWS_PER;
    int r1 = r0 + ROWS_PER;
    if (r1 > B_ROWS) r1 = B_ROWS;

    const v4i st = *(const v4i*)(starts + s4);

    // Two fp32 accumulators per lane -> exactly the 16x4 fp32 A-tile of a wave.
    float a0 = 0.0f, a1 = 0.0f;
    for (int j = r0; j < r1; ++j) {
        const size_t off = (size_t)j * S_COLS + (size_t)s4;
        v4f p = __builtin_nontemporal_load((const v4f*)(pred + off));
        v4f a = __builtin_nontemporal_load((const v4f*)(act + off));
        v4f d = p - a;
        float d0 = (j >= st.x) ? d.x : 0.0f;
        float d1 = (j >= st.y) ? d.y : 0.0f;
        float d2 = (j >= st.z) ? d.z : 0.0f;
        float d3 = (j >= st.w) ? d.w : 0.0f;
        a0 = fmaf(d0, d0, a0);
        a0 = fmaf(d1, d1, a0);
        a1 = fmaf(d2, d2, a1);
        a1 = fmaf(d3, d3, a1);
    }

    // Wave-level reduction on the matrix pipe: D = A(16x4) * ones(4x16).
    // Every D row is the row-sum of A; summing a lane's 8 D VGPRs leaves the
    // half-wave total, one shfl_xor(16) completes the 64-value reduction.
    v2f A; A.x = a0; A.y = a1;
    v2f Bones; Bones.x = 1.0f; Bones.y = 1.0f;
    v8f C = {};
    C = __builtin_amdgcn_wmma_f32_16x16x4_f32(
        /*neg_a=*/false, A, /*neg_b=*/false, Bones,
        /*c_mod=*/(short)0, C, /*reuse_a=*/false, /*reuse_b=*/false);
    float w = 0.0f;
#pragma unroll
    for (int i = 0; i < 8; ++i) w += C[i];
    w += __shfl_xor(w, 16, 32);          // wave32: combine the two half-waves

    __shared__ float wsum[TPB / 32];
    const int lane = tid & 31, wid = tid >> 5;
    if (lane == 0) wsum[wid] = w;
    __syncthreads();
    if (tid == 0) {
        float t = 0.0f;
#pragma unroll
        for (int i = 0; i < TPB / 32; ++i) t += wsum[i];   // fixed order
        partials[blockIdx.y * GX + blockIdx.x] = t;
    }
}

// --- Kernel 3: final reduce + exact int32 count + sqrt ---------------------
__global__ __launch_bounds__(TPB) void rmse_final(const float* __restrict__ partials,
                                                  const int*   __restrict__ starts,
                                                  float*       __restrict__ out) {
    const int tid = threadIdx.x;
    float s = 0.0f;
    for (int i = tid; i < NPART; i += TPB) s += partials[i];
    // count = sum_s (B - starts[s]); max ~31.8M > 2^24 -> keep it in int32.
    int c = 0;
    for (int i = tid; i < S_COLS; i += TPB) c += (B_ROWS - starts[i]);

    __shared__ float sf[TPB];
    __shared__ int   si[TPB];
    sf[tid] = s; si[tid] = c;
    __syncthreads();
    for (int o = TPB / 2; o > 0; o >>= 1) {
        if (tid < o) { sf[tid] += sf[tid + o]; si[tid] += si[tid + o]; }
        __syncthreads();
    }
    if (tid == 0) out[0] = sqrtf(sf[0] / (float)si[0]);
}

extern "C" void kernel_launch(void* const* d_in, const int* in_sizes, int n_in,
                              void* d_out, int out_size, void* d_ws, size_t ws_size,
                              hipStream_t stream) {
    const float* pred = (const float*)d_in[0];   // [1941, 16384] f32
    const float* act  = (const float*)d_in[1];   // [1941, 16384] f32
    const int*   idx  = (const int*)d_in[2];     // [16384] i32
    const int*   rvsi = (const int*)d_in[3];     // [30490] i32
    float* out = (float*)d_out;

    int*   starts   = (int*)d_ws;                                 // 64 KB
    float* partials = (float*)((char*)d_ws + S_COLS * sizeof(int)); // 8 KB

    rmse_prep<<<S_COLS / TPB, TPB, 0, stream>>>(idx, rvsi, starts);
    dim3 grid(GX, ROW_CHUNKS);
    rmse_partial<<<grid, TPB, 0, stream>>>(pred, act, starts, partials);
    rmse_final<<<1, TPB, 0, stream>>>(partials, starts, out);
}